// GATBlock_90443421319378
// MI455X (gfx1250) — compile-verified
//
#include <hip/hip_runtime.h>

typedef __attribute__((ext_vector_type(2))) float v2f;
typedef __attribute__((ext_vector_type(8))) float v8f;

#define B_DIM   8
#define N_DIM   2048
#define FIN     128
#define FOUT    64
#define SLOPE   0.2f

__device__ __forceinline__ float lrelu(float t) {
    // 0 < slope < 1  =>  leaky_relu(t) == max(t, slope*t)
    return fmaxf(t, SLOPE * t);
}

// ---------------------------------------------------------------------------
// Kernel 1: h = x @ W   (per batch: 2048x128 @ 128x64 -> 2048x64)
// grid = 256 blocks (8 batches * 32 row-tiles of 64 rows), block = 128 (4 waves)
// Each wave computes one 16x64 tile: 4 N-tiles, K loop 128 in steps of 4.
// ---------------------------------------------------------------------------
__global__ void __launch_bounds__(128)
gat_gemm_xw(const float* __restrict__ x, const float* __restrict__ W,
            float* __restrict__ h)
{
    __shared__ float lds_w[FIN * FOUT];           // 32 KB
    const int tid  = threadIdx.x;
    const int blk  = blockIdx.x;
    const int b    = blk >> 5;                                  // batch
    const int row0 = ((blk & 31) << 6) + ((tid >> 5) << 4);     // tile row base
    const int lane = tid & 31;
    const int half = lane >> 4;
    const int mrow = lane & 15;

    for (int t = tid; t < FIN * FOUT; t += 128)
        lds_w[t] = W[t];
    __syncthreads();

    const float* xrow = x + ((size_t)b * N_DIM + (row0 + mrow)) * FIN;

    v8f acc[4] = {};

    for (int k = 0; k < FIN; k += 4) {
        const int k0 = k + 2 * half;              // K index pair for this lane
        v2f a = *(const v2f*)(xrow + k0);         // A[m=mrow][k0], A[m][k0+1]
        #pragma unroll
        for (int nt = 0; nt < 4; ++nt) {
            v2f bb;
            bb.x = lds_w[ k0      * FOUT + nt * 16 + mrow];  // B[k0  ][n]
            bb.y = lds_w[(k0 + 1) * FOUT + nt * 16 + mrow];  // B[k0+1][n]
            acc[nt] = __builtin_amdgcn_wmma_f32_16x16x4_f32(
                false, a, false, bb, (short)0, acc[nt], false, false);
        }
    }

    float* hb = h + (size_t)b * N_DIM * FOUT;
    #pragma unroll
    for (int v = 0; v < 8; ++v) {
        const int row = row0 + v + 8 * half;      // C layout: M = v + 8*half
        #pragma unroll
        for (int nt = 0; nt < 4; ++nt)
            hb[row * FOUT + nt * 16 + mrow] = acc[nt][v];
    }
}

// ---------------------------------------------------------------------------
// Kernel 2: s1[b,i] = h[b,i,:] . a[:64] ,  s2[b,i] = h[b,i,:] . a[64:128]
// grid = 64 blocks * 256 threads, one thread per (b,i)
// ---------------------------------------------------------------------------
__global__ void __launch_bounds__(256)
gat_scores(const float* __restrict__ h, const float* __restrict__ a,
           float* __restrict__ s1, float* __restrict__ s2)
{
    const int gid = blockIdx.x * 256 + threadIdx.x;   // 0 .. B*N-1
    const float* hr = h + (size_t)gid * FOUT;
    float r1 = 0.f, r2 = 0.f;
    #pragma unroll
    for (int f = 0; f < FOUT; ++f) {
        const float hv = hr[f];
        r1 += hv * a[f];            // uniform -> scalar loads
        r2 += hv * a[FOUT + f];
    }
    s1[gid] = r1;
    s2[gid] = r2;
}

// ---------------------------------------------------------------------------
// Kernel 3: per-column softmax constant  c[b,j] = m[b,j] + log(D[b,j])
//   m[j] = max_i lrelu(s1[i]+s2[j]),  D[j] = sum_i exp(lrelu(s1[i]+s2[j]) - m[j])
// Then attention[i,j] = exp(lrelu(s1[i]+s2[j]) - c[j]) exactly.
// grid = 64 blocks * 256 threads; block handles one batch's 256-j chunk.
// ---------------------------------------------------------------------------
__global__ void __launch_bounds__(256)
gat_colstats(const float* __restrict__ s1, const float* __restrict__ s2,
             float* __restrict__ cexp)
{
    __shared__ float lds_s1[N_DIM];               // 8 KB
    const int b = blockIdx.x >> 3;
    const int j = ((blockIdx.x & 7) << 8) + threadIdx.x;
    const float* s1b = s1 + b * N_DIM;

    for (int t = threadIdx.x; t < N_DIM; t += 256)
        lds_s1[t] = s1b[t];
    __syncthreads();

    const float s2j = s2[b * N_DIM + j];
    float m = -3.4e38f;
    for (int i = 0; i < N_DIM; ++i)
        m = fmaxf(m, lrelu(lds_s1[i] + s2j));
    float D = 0.f;
    for (int i = 0; i < N_DIM; ++i)
        D += __expf(lrelu(lds_s1[i] + s2j) - m);
    cexp[b * N_DIM + j] = m + __logf(D);
}

// ---------------------------------------------------------------------------
// Kernel 4 (fused): out = lrelu( attention @ h )
//   out[b,i,f] = lrelu( sum_j exp(lrelu(s1[i]+s2[j]) - c[j]) * h[b,j,f] )
// grid = 128 blocks (8 batches * 16 i-blocks of 128 rows), block = 256 (8 waves)
// Each wave owns one 16-row tile; K(=j) dimension streamed in 128-chunks via LDS.
// ---------------------------------------------------------------------------
__global__ void __launch_bounds__(256)
gat_attn_out(const float* __restrict__ h, const float* __restrict__ s1,
             const float* __restrict__ s2, const float* __restrict__ cexp,
             float* __restrict__ out)
{
    __shared__ float lds_h[128 * FOUT];           // 32 KB: h[j0..j0+127, 0..63]
    __shared__ float lds_s2[128];
    __shared__ float lds_c[128];

    const int b    = blockIdx.x >> 4;
    const int i0   = ((blockIdx.x & 15) << 7) + ((threadIdx.x >> 5) << 4);
    const int lane = threadIdx.x & 31;
    const int half = lane >> 4;
    const int mrow = lane & 15;

    const float  s1row = s1[b * N_DIM + i0 + mrow];   // A-row value per lane
    const float* hb    = h + (size_t)b * N_DIM * FOUT;

    v8f acc[4] = {};

    for (int j0 = 0; j0 < N_DIM; j0 += 128) {
        __syncthreads();
        for (int t = threadIdx.x; t < 128 * FOUT; t += 256)
            lds_h[t] = hb[j0 * FOUT + t];
        if (threadIdx.x < 128) {
            lds_s2[threadIdx.x] = s2  [b * N_DIM + j0 + threadIdx.x];
            lds_c [threadIdx.x] = cexp[b * N_DIM + j0 + threadIdx.x];
        }
        if (j0 + 128 < N_DIM)   // warm L2/WGP$ for the next chunk
            __builtin_prefetch(hb + (j0 + 128) * FOUT + threadIdx.x * 32, 0, 1);
        __syncthreads();

        for (int jj = 0; jj < 128; jj += 4) {
            const int k0 = jj + 2 * half;
            v2f a;     // attention weights built on the fly (one exp each)
            a.x = __expf(lrelu(s1row + lds_s2[k0    ]) - lds_c[k0    ]);
            a.y = __expf(lrelu(s1row + lds_s2[k0 + 1]) - lds_c[k0 + 1]);
            #pragma unroll
            for (int nt = 0; nt < 4; ++nt) {
                v2f bb;
                bb.x = lds_h[ k0      * FOUT + nt * 16 + mrow];
                bb.y = lds_h[(k0 + 1) * FOUT + nt * 16 + mrow];
                acc[nt] = __builtin_amdgcn_wmma_f32_16x16x4_f32(
                    false, a, false, bb, (short)0, acc[nt], false, false);
            }
        }
    }

    float* ob = out + (size_t)b * N_DIM * FOUT;
    #pragma unroll
    for (int v = 0; v < 8; ++v) {
        const int row = i0 + v + 8 * half;
        #pragma unroll
        for (int nt = 0; nt < 4; ++nt)
            ob[row * FOUT + nt * 16 + mrow] = lrelu(acc[nt][v]);
    }
}

// ---------------------------------------------------------------------------
extern "C" void kernel_launch(void* const* d_in, const int* in_sizes, int n_in,
                              void* d_out, int out_size, void* d_ws, size_t ws_size,
                              hipStream_t stream) {
    const float* x = (const float*)d_in[0];   // (8, 2048, 128)
    const float* W = (const float*)d_in[1];   // (128, 64)
    const float* a = (const float*)d_in[2];   // (128, 1)
    float* out = (float*)d_out;               // (8, 2048, 64)

    // Workspace layout (floats): h | s1 | s2 | c
    float* ws   = (float*)d_ws;
    float* h    = ws;                                   // 8*2048*64 = 1,048,576
    float* s1   = ws + (size_t)B_DIM * N_DIM * FOUT;    // 16,384
    float* s2   = s1 + (size_t)B_DIM * N_DIM;           // 16,384
    float* cexp = s2 + (size_t)B_DIM * N_DIM;           // 16,384

    // K1: h = x @ W
    gat_gemm_xw<<<dim3(B_DIM * (N_DIM / 64)), dim3(128), 0, stream>>>(x, W, h);
    // K2: s1, s2
    gat_scores<<<dim3((B_DIM * N_DIM) / 256), dim3(256), 0, stream>>>(h, a, s1, s2);
    // K3: per-column softmax constants
    gat_colstats<<<dim3((B_DIM * N_DIM) / 256), dim3(256), 0, stream>>>(s1, s2, cexp);
    // K4: fused attention @ h + lrelu
    gat_attn_out<<<dim3(B_DIM * (N_DIM / 128)), dim3(256), 0, stream>>>(h, s1, s2, cexp, out);
}